// CliffordPC_13125420057230
// MI455X (gfx1250) — compile-verified
//
#include <hip/hip_runtime.h>

typedef float v2f __attribute__((ext_vector_type(2)));
typedef float v4f __attribute__((ext_vector_type(4)));
typedef float v8f __attribute__((ext_vector_type(8)));
typedef int   v4i __attribute__((ext_vector_type(4)));

#define ALPHA 0.01f
#define NITER 20
#define KT 32
#define LDS_STRIDE 36              // 32 + 4 pad: rows 16B-aligned, conflict-free banks
#define TILE_FLOATS (128 * LDS_STRIDE)

#define AS1 __attribute__((address_space(1)))
#define AS3 __attribute__((address_space(3)))

#if __has_builtin(__builtin_amdgcn_global_load_async_to_lds_b128)
#define USE_ASYNC 1
#else
#define USE_ASYNC 0
#endif

#if USE_ASYNC
#if __has_builtin(__builtin_amdgcn_s_wait_asynccnt)
#define WAIT_ASYNC(n) __builtin_amdgcn_s_wait_asynccnt(n)
#else
#define WAIT_ASYNC(n) asm volatile("s_wait_asynccnt %0" ::"n"(n) : "memory")
#endif
#endif

// ---------------------------------------------------------------------------
// Clifford signs (SIG_G = [1,1,1] -> metric factors are all +1)
// ---------------------------------------------------------------------------
__device__ __forceinline__ float gp_sign(int a, int b) {
    int s = 0;
    int aa = a >> 1;
    while (aa) { s += __popc(aa & b); aa >>= 1; }
    return (s & 1) ? -1.0f : 1.0f;
}
__device__ __forceinline__ float rev_sign(int c) {
    int k = __popc(c);
    return ((k * (k - 1) / 2) & 1) ? -1.0f : 1.0f;
}

// ---------------------------------------------------------------------------
// Expand W [D0, D1, 8] into a TRANSPOSED dense GEMM operand BT [N, K]:
//   rev==0: BT[(i*8+c) * (D1*8) + (j*8+a)] = sign(a,a^c) * W[i,j,a^c]
//   rev==1: BT[(p*8+c) * (D0*8) + (q*8+a)] = sign(a,a^c)*REV[a^c]*W[q,p,a^c]
// Storing [N,K] makes the B tile identical in shape/stride to the A tile, so
// both stage with b128 copies and fragment-load as ds_load_b64.
// ---------------------------------------------------------------------------
__global__ void expand_w_kernel(const float* __restrict__ W, float* __restrict__ BT,
                                int D0, int D1, int rev) {
    int Dm = rev ? D1 : D0;          // M-cols dimension /8
    int Dk = rev ? D0 : D1;          // K-rows dimension /8
    long total = (long)D0 * D1 * 64;
    long idx = (long)blockIdx.x * blockDim.x + threadIdx.x;
    if (idx >= total) return;
    int c = (int)(idx & 7);
    int a = (int)((idx >> 3) & 7);
    long rest = idx >> 6;
    int mIdx = (int)(rest % Dm);
    int kIdx = (int)(rest / Dm);
    int b = a ^ c;
    float s = gp_sign(a, b);
    if (rev) s *= rev_sign(b);
    size_t widx = rev ? ((size_t)kIdx * D1 + mIdx) * 8 + b
                      : ((size_t)mIdx * D1 + kIdx) * 8 + b;
    BT[(size_t)(mIdx * 8 + c) * (Dk * 8) + (kIdx * 8 + a)] = s * W[widx];
}

// ---------------------------------------------------------------------------
// Tile staging: 128 rows x 32 floats from row-major [rows, K] global into LDS
// with padded stride. 1024 x b128 segments, 4 per thread, coalesced.
// ---------------------------------------------------------------------------
#if USE_ASYNC
__device__ __forceinline__ void stage_tile(const float* __restrict__ g, float* l,
                                           int K, int tid) {
#pragma unroll
    for (int i = 0; i < 4; ++i) {
        int seg = tid + 256 * i;
        int row = seg >> 3;
        int c4 = (seg & 7) * 4;
        const float* gp = g + (size_t)row * K + c4;
        float* lp = l + row * LDS_STRIDE + c4;
        __builtin_amdgcn_global_load_async_to_lds_b128(
            (AS1 v4i*)gp, (AS3 v4i*)lp, 0, 0);
    }
}
#else
__device__ __forceinline__ void stage_tile(const float* __restrict__ g, float* l,
                                           int K, int tid) {
#pragma unroll
    for (int i = 0; i < 4; ++i) {
        int seg = tid + 256 * i;
        int row = seg >> 3;
        int c4 = (seg & 7) * 4;
        v4f v = *(const v4f*)(g + (size_t)row * K + c4);
        *(v4f*)(l + row * LDS_STRIDE + c4) = v;
    }
}
#endif

// ---------------------------------------------------------------------------
// fp32 WMMA GEMM, C = A[M,K] x B[K,N] with B given transposed (BT [N,K]).
// 128x128 block tile, 8 waves, wave tile 64x32 = 4x2 subtiles of 16x16.
// LDS double-buffered KT=32 stages; fragments via ds_load_b64.
// Fused epilogues:
//   MODE 0: outA = X - C                         (E0 = x - pred_prev)
//   MODE 1: outA = C                             (P2)
//   MODE 2: g=clip(H-P2-C); Hn=H-a*g; outA=Hn; outB=Hn-P2
//   MODE 3: outA = H + a*clip(C)
// ---------------------------------------------------------------------------
__device__ __forceinline__ v8f wmma4(v2f a, v2f b, v8f c) {
    return __builtin_amdgcn_wmma_f32_16x16x4_f32(
        false, a, false, b, (short)0, c, false, false);
}

template <int MODE>
__global__ __launch_bounds__(256) void gemm_ep(
    const float* __restrict__ A, const float* __restrict__ BT,
    int M, int N, int K,
    const float* __restrict__ X,
    const float* __restrict__ P2,
    const float* __restrict__ H,
    float* __restrict__ outA,
    float* __restrict__ outB) {

    __shared__ float sA[2][TILE_FLOATS];
    __shared__ float sB[2][TILE_FLOATS];

    const int tid = threadIdx.x;
    const int wave = tid >> 5;
    const int lane = tid & 31;
    const int lr = lane & 15;            // 0..15
    const int lh = lane >> 4;            // 0..1
    const int wrow = wave >> 2;          // 0..1
    const int wcol = wave & 3;           // 0..3

    const int rowbase = blockIdx.x * 128 + wrow * 64;
    const int colbase = blockIdx.y * 128 + wcol * 32;

    const float* Ablk = A + (size_t)blockIdx.x * 128 * K;
    const float* Bblk = BT + (size_t)blockIdx.y * 128 * K;

    v8f acc[4][2];
#pragma unroll
    for (int t = 0; t < 4; ++t)
#pragma unroll
        for (int s = 0; s < 2; ++s)
            acc[t][s] = (v8f){0.f, 0.f, 0.f, 0.f, 0.f, 0.f, 0.f, 0.f};

    const int nst = K / KT;

    stage_tile(Ablk, &sA[0][0], K, tid);
    stage_tile(Bblk, &sB[0][0], K, tid);

    for (int st = 0; st < nst; ++st) {
        const int cur = st & 1;
        if (st + 1 < nst) {
            stage_tile(Ablk + (st + 1) * KT, &sA[cur ^ 1][0], K, tid);
            stage_tile(Bblk + (st + 1) * KT, &sB[cur ^ 1][0], K, tid);
#if USE_ASYNC
            WAIT_ASYNC(8);     // current buffer's 8 copies complete (in-order)
#endif
        } else {
#if USE_ASYNC
            WAIT_ASYNC(0);
#endif
        }
        __syncthreads();

        // Fragment bases (byte offsets below are compile-time immediates).
        // A frag (16x4 MxK): element(m,k) -> lane (k/2)*16+m, vgpr k%2
        // B frag (4x16 KxN): element(k,n) -> lane (k/2)*16+n, vgpr k%2
        const float* aB = &sA[cur][0] + (wrow * 64 + lr) * LDS_STRIDE + 2 * lh;
        const float* bB = &sB[cur][0] + (wcol * 32 + lr) * LDS_STRIDE + 2 * lh;

#pragma unroll
        for (int kb = 0; kb < KT; kb += 4) {
            v2f af[4];
#pragma unroll
            for (int t = 0; t < 4; ++t)
                af[t] = *(const v2f*)(aB + t * 16 * LDS_STRIDE + kb);
            v2f bf[2];
#pragma unroll
            for (int s = 0; s < 2; ++s)
                bf[s] = *(const v2f*)(bB + s * 16 * LDS_STRIDE + kb);
#pragma unroll
            for (int t = 0; t < 4; ++t)
#pragma unroll
                for (int s = 0; s < 2; ++s)
                    acc[t][s] = wmma4(af[t], bf[s], acc[t][s]);
        }
        __syncthreads();
    }

    // Epilogue. C/D layout: vgpr r holds m = r + 8*(lane/16), n = lane%16.
#pragma unroll
    for (int t = 0; t < 4; ++t) {
#pragma unroll
        for (int s = 0; s < 2; ++s) {
#pragma unroll
            for (int r = 0; r < 8; ++r) {
                const int m = rowbase + t * 16 + r + 8 * lh;
                const int n = colbase + s * 16 + lr;
                const size_t o = (size_t)m * N + n;
                const float c = acc[t][s][r];
                if (MODE == 0) {
                    outA[o] = X[o] - c;
                } else if (MODE == 1) {
                    outA[o] = c;
                } else if (MODE == 2) {
                    const float h = H[o];
                    const float p = P2[o];
                    float g = h - p - c;                    // grad = -C + (H - P2)
                    g = fminf(1.0f, fmaxf(-1.0f, g));
                    const float hn = h - ALPHA * g;
                    outA[o] = hn;                           // new H1
                    outB[o] = hn - p;                       // E1
                } else {                                    // MODE 3
                    float g = fminf(1.0f, fmaxf(-1.0f, c)); // clip(-C) = -clip(C)
                    outA[o] = H[o] + ALPHA * g;
                }
            }
        }
    }
}

// ---------------------------------------------------------------------------
// Orchestration
// ---------------------------------------------------------------------------
extern "C" void kernel_launch(void* const* d_in, const int* in_sizes, int n_in,
                              void* d_out, int out_size, void* d_ws, size_t ws_size,
                              hipStream_t stream) {
    (void)in_sizes; (void)n_in; (void)out_size; (void)ws_size;

    const float* x  = (const float*)d_in[0];   // [4096, 256, 8]
    const float* W1 = (const float*)d_in[1];   // [256, 128, 8]
    const float* W2 = (const float*)d_in[2];   // [128, 64, 8]
    const float* h1 = (const float*)d_in[3];   // [4096, 128, 8]
    const float* h2 = (const float*)d_in[4];   // [4096, 64, 8]
    float* out = (float*)d_out;

    const int Bn = 4096;
    const int N0 = 256 * 8;   // 2048
    const int N1 = 128 * 8;   // 1024
    const int N2 = 64 * 8;    // 512

    float* ws = (float*)d_ws;
    size_t off = 0;
    float* Wexp1  = ws + off; off += (size_t)N1 * N0;   // BT: [2048, 1024]
    float* WexpR1 = ws + off; off += (size_t)N0 * N1;   // BT: [1024, 2048]
    float* Wexp2  = ws + off; off += (size_t)N2 * N1;   // BT: [1024, 512]
    float* WexpR2 = ws + off; off += (size_t)N1 * N2;   // BT: [512, 1024]
    float* E0 = ws + off; off += (size_t)Bn * N0;       // [4096, 2048]
    float* P2 = ws + off; off += (size_t)Bn * N1;       // [4096, 1024]
    float* E1 = ws + off; off += (size_t)Bn * N1;       // [4096, 1024]
    float* H1 = ws + off; off += (size_t)Bn * N1;       // [4096, 1024]
    float* H2 = ws + off; off += (size_t)Bn * N2;       // [4096, 512]

    (void)hipMemcpyAsync(H1, h1, (size_t)Bn * N1 * sizeof(float), hipMemcpyDeviceToDevice, stream);
    (void)hipMemcpyAsync(H2, h2, (size_t)Bn * N2 * sizeof(float), hipMemcpyDeviceToDevice, stream);

    {
        long t1 = 256L * 128 * 64;
        long t2 = 128L * 64 * 64;
        expand_w_kernel<<<(unsigned)((t1 + 255) / 256), 256, 0, stream>>>(W1, Wexp1,  256, 128, 0);
        expand_w_kernel<<<(unsigned)((t1 + 255) / 256), 256, 0, stream>>>(W1, WexpR1, 256, 128, 1);
        expand_w_kernel<<<(unsigned)((t2 + 255) / 256), 256, 0, stream>>>(W2, Wexp2,  128, 64, 0);
        expand_w_kernel<<<(unsigned)((t2 + 255) / 256), 256, 0, stream>>>(W2, WexpR2, 128, 64, 1);
    }

    for (int it = 0; it < NITER; ++it) {
        // E0 = x - H1 x Wexp1            [4096,1024] x [1024,2048]
        gemm_ep<0><<<dim3(Bn / 128, N0 / 128), 256, 0, stream>>>(
            H1, Wexp1, Bn, N0, N1, x, nullptr, nullptr, E0, nullptr);
        // P2 = H2 x Wexp2                [4096,512] x [512,1024]
        gemm_ep<1><<<dim3(Bn / 128, N1 / 128), 256, 0, stream>>>(
            H2, Wexp2, Bn, N1, N2, nullptr, nullptr, nullptr, P2, nullptr);
        // G1 = E0 x WexpR1; H1 update; E1 = H1new - P2
        gemm_ep<2><<<dim3(Bn / 128, N1 / 128), 256, 0, stream>>>(
            E0, WexpR1, Bn, N1, N0, nullptr, P2, H1, H1, E1);
        // G2 = E1 x WexpR2; H2 update
        gemm_ep<3><<<dim3(Bn / 128, N2 / 128), 256, 0, stream>>>(
            E1, WexpR2, Bn, N2, N1, nullptr, nullptr, H2, H2, nullptr);
    }

    size_t xo = (size_t)Bn * N0;
    size_t h1o = (size_t)Bn * N1;
    size_t h2o = (size_t)Bn * N2;
    (void)hipMemcpyAsync(out,            x,  xo  * sizeof(float), hipMemcpyDeviceToDevice, stream);
    (void)hipMemcpyAsync(out + xo,       H1, h1o * sizeof(float), hipMemcpyDeviceToDevice, stream);
    (void)hipMemcpyAsync(out + xo + h1o, H2, h2o * sizeof(float), hipMemcpyDeviceToDevice, stream);
}